// GroupTopK_24592982737273
// MI455X (gfx1250) — compile-verified
//
#include <hip/hip_runtime.h>

// DeepSeek noaux-tc grouped top-k routing, MI455X (gfx1250, wave32).
// Bandwidth-bound: 128MB read / 8MB write -> ~5.8us floor @ 23.3 TB/s.
// One wave32 per token; lane L owns experts [8L, 8L+8). Group g (32 experts)
// == lane quad [4g, 4g+4).
//
// Selection engine: each lane sorts its 8 keys (mono(score)<<32 | (7-j))
// descending ONCE (Batcher 19-CE network). Then each of the 8 extractions is
// just: butterfly-max over the 32 sorted heads' hi-dwords + ballot owner
// (lowest lane == lowest expert index, exact jax.lax.top_k tie-break; within
// a lane the smallest j is at the head by key construction) + a 7-element
// shift-pop on the owner. Winner weights broadcast by one uniform-address
// ds_load_b32 from a wave-private LDS row of exact sigmoids (same-wave DS
// ops are in-order -> no barriers). No scratch, no dynamic reg indexing.

#define NE        256   // experts
#define EPL       8     // experts per lane (256/32)
#define NGRP      8     // groups
#define TOPKG     4     // groups kept
#define TOPK      8     // experts kept
#define TPW       4     // tokens per wave (gives prefetch a target)
#define WAVES     8     // waves per block (256 threads)

__device__ __forceinline__ unsigned mono_bits(float f) {
    // Order-preserving float->uint transform (no NaNs in this workload).
    unsigned u = __float_as_uint(f);
    return u ^ (unsigned)(((int)u >> 31) | 0x80000000);
}

__device__ __forceinline__ float unmono_bits(unsigned m) {
    // Inverse of mono_bits.
    unsigned b = m ^ (0x80000000u | (unsigned)~((int)m >> 31));
    return __uint_as_float(b);
}

__device__ __forceinline__ float readlane_f32(float v, int lane) {
    return __uint_as_float(__builtin_amdgcn_readlane(__float_as_uint(v), lane));
}

__global__ __launch_bounds__(WAVES * 32)
void GroupTopK_routing_kernel(const float* __restrict__ scores,
                              const float* __restrict__ bias,
                              const float* __restrict__ scale_p,
                              const int*   __restrict__ renorm_p,
                              float*       __restrict__ out_w,
                              int*         __restrict__ out_i,
                              int T)
{
    // Wave-private row of exact unbiased sigmoid values (8KB/block).
    __shared__ __align__(16) float s_sig[WAVES * NE];

    const int lane = threadIdx.x & 31;
    const int wave = threadIdx.x >> 5;
    const int t0   = (blockIdx.x * WAVES + wave) * TPW;
    float* srow    = &s_sig[wave * NE];

    // Per-lane correction bias for experts 8*lane .. 8*lane+7 (loop-invariant).
    float4 bv0 = ((const float4*)bias)[lane * 2 + 0];
    float4 bv1 = ((const float4*)bias)[lane * 2 + 1];
    float bb[EPL] = {bv0.x, bv0.y, bv0.z, bv0.w, bv1.x, bv1.y, bv1.z, bv1.w};

    const float scale  = scale_p[0];
    const int   renorm = renorm_p[0];

    for (int it = 0; it < TPW; ++it) {
        const int t = t0 + it;
        if (t >= T) break;

        const float4* row = (const float4*)(scores + (size_t)t * NE);

        // CDNA5 prefetch (global_prefetch_b8): pull next token's 1KB row
        // toward the caches while we do this token's selection math.
        if (it + 1 < TPW && t + 1 < T) {
            const float4* nrow = (const float4*)(scores + (size_t)(t + 1) * NE);
            __builtin_prefetch((const void*)(nrow + lane * 2), 0, 0);
        }

        // ---- load scores: 2 x global_load_b128 per lane, coalesced 1KB/wave
        float4 x0 = row[lane * 2 + 0];
        float4 x1 = row[lane * 2 + 1];
        float xs[EPL] = {x0.x, x0.y, x0.z, x0.w, x1.x, x1.y, x1.z, x1.w};

        float sv[EPL];   // unbiased sigmoid (weights come from these)
        unsigned long long kk[EPL];  // (mono(sb)<<32) | (7-j)
        #pragma unroll
        for (int j = 0; j < EPL; ++j) {
            float s = 1.0f / (1.0f + __expf(-xs[j]));
            sv[j] = s;
            float sb = s + bb[j];
            kk[j] = ((unsigned long long)mono_bits(sb) << 32)
                  | (unsigned)(EPL - 1 - j);
        }

        // Stage exact sigmoids into the wave-private LDS row (2x b128 store).
        float4 w0; w0.x = sv[0]; w0.y = sv[1]; w0.z = sv[2]; w0.w = sv[3];
        float4 w1; w1.x = sv[4]; w1.y = sv[5]; w1.z = sv[6]; w1.w = sv[7];
        ((float4*)&srow[lane * EPL])[0] = w0;
        ((float4*)&srow[lane * EPL])[1] = w1;

        // ---- per-lane descending sort: Batcher odd-even merge, 19 CEs ----
        #define CE_(i_, j_) { unsigned long long a_ = kk[i_], b_ = kk[j_]; \
                              bool lt_ = a_ < b_;                          \
                              kk[i_] = lt_ ? b_ : a_;                      \
                              kk[j_] = lt_ ? a_ : b_; }
        CE_(0,1) CE_(2,3) CE_(4,5) CE_(6,7)
        CE_(0,2) CE_(1,3) CE_(4,6) CE_(5,7)
        CE_(1,2) CE_(5,6)
        CE_(0,4) CE_(1,5) CE_(2,6) CE_(3,7)
        CE_(2,4) CE_(3,5)
        CE_(1,2) CE_(3,4) CE_(5,6)
        #undef CE_

        // ---- group score = sum of top-2 biased scores within the group ----
        // Lane-local top-2 are the sorted heads; quad merge in mono domain.
        unsigned a1 = (unsigned)(kk[0] >> 32);
        unsigned a2 = (unsigned)(kk[1] >> 32);
        #pragma unroll
        for (int d = 1; d <= 2; d <<= 1) {
            unsigned o1 = __shfl_xor(a1, d, 32);
            unsigned o2 = __shfl_xor(a2, d, 32);
            bool gt = o1 > a1;
            unsigned n1 = gt ? o1 : a1;
            unsigned n2 = gt ? (a1 > o2 ? a1 : o2) : (a2 > o1 ? a2 : o1);
            a1 = n1; a2 = n2;
        }
        const float gscore = unmono_bits(a1) + unmono_bits(a2);

        // ---- top-4 groups of 8 (tie-break: smaller group index) ----------
        const int myg = lane >> 2;
        int rank = 0;
        #pragma unroll
        for (int g = 0; g < NGRP; ++g) {
            float og = readlane_f32(gscore, g * 4);
            rank += (og > gscore) || (og == gscore && g < myg);
        }
        const bool selected = (rank < TOPKG);

        // Mask unselected groups: zero the hi (mono) halves. mono of any
        // finite float >= 0x00800000 > 0, so masked heads can never win.
        #pragma unroll
        for (int j = 0; j < EPL; ++j)
            kk[j] = selected ? kk[j] : (kk[j] & 0xFFFFFFFFull);

        // ---- iterative top-8: heads-only tournament -----------------------
        float wsum = 0.0f;
        float myw  = 0.0f;
        int   myid = 0;
        #pragma unroll
        for (int k = 0; k < TOPK; ++k) {
            const unsigned h0 = (unsigned)(kk[0] >> 32);   // my head (free)

            unsigned best = h0;
            #pragma unroll
            for (int d = 16; d >= 1; d >>= 1) {
                unsigned o = __shfl_xor(best, d, 32);
                best = o > best ? o : best;
            }

            // owner = lowest lane whose head equals the global max
            unsigned bal    = (unsigned)__ballot(h0 == best);
            const int owner = __ffs(bal) - 1;                 // uniform

            // winner slot from the owner's head low dword (scalar math)
            const int jj  = (EPL - 1) -
                (__builtin_amdgcn_readlane((int)(unsigned)kk[0], owner) & 7);
            const int idx = owner * EPL + jj;

            // weight = exact unbiased sigmoid, uniform-address LDS broadcast
            const float w = srow[idx];

            wsum += w;                          // uniform across lanes
            myw  = (lane == k) ? w   : myw;
            myid = (lane == k) ? idx : myid;

            // consume: shift-pop the owner's sorted list
            const bool am = (lane == owner);
            #pragma unroll
            for (int j = 0; j < EPL - 1; ++j)
                kk[j] = am ? kk[j + 1] : kk[j];
            kk[EPL - 1] = am ? 0ull : kk[EPL - 1];
        }

        const float f = renorm ? (1.0f / (wsum + 1e-20f)) : 1.0f;
        if (lane < TOPK) {
            out_w[(size_t)t * TOPK + lane] = myw * f * scale;
            out_i[(size_t)t * TOPK + lane] = myid;
        }
    }
}

extern "C" void kernel_launch(void* const* d_in, const int* in_sizes, int n_in,
                              void* d_out, int out_size, void* d_ws, size_t ws_size,
                              hipStream_t stream) {
    // setup_inputs order:
    //   0: scores [T,256] f32   1: correction_bias [256] f32
    //   2: routed_scaling_factor f32[1]
    //   3: n_group  4: topk_group  5: topk  6: renormalize (int[1] each)
    const float* scores = (const float*)d_in[0];
    const float* bias   = (const float*)d_in[1];
    const float* scale  = (const float*)d_in[2];
    const int*   renorm = (const int*)d_in[6];

    const int T = in_sizes[0] / NE;

    // Output tuple concatenated flat: weights [T,8] f32, then ids [T,8] i32
    float* out_w = (float*)d_out;
    int*   out_i = (int*)((float*)d_out + (size_t)T * TOPK);

    const int tokens_per_block = WAVES * TPW;           // 32 tokens / block
    const int grid = (T + tokens_per_block - 1) / tokens_per_block;

    GroupTopK_routing_kernel<<<grid, WAVES * 32, 0, stream>>>(
        scores, bias, scale, renorm, out_w, out_i, T);
}